// MultiHeadAttentionRoPE_48335561949777
// MI455X (gfx1250) — compile-verified
//
#include <hip/hip_runtime.h>
#include <hip/hip_bf16.h>

// MI455X / gfx1250, wave32. All matmuls on the matrix pipe via
// V_WMMA_F32_16X16X4_F32 (exact fp32 — reference is fp32).
// Attention stages K/V tiles into LDS with the Tensor Data Mover
// (tensor_load_to_lds + s_wait_tensorcnt), double-buffered.

typedef __attribute__((ext_vector_type(2))) float v2f;
typedef __attribute__((ext_vector_type(8))) float v8f;
typedef __attribute__((ext_vector_type(4))) unsigned int u32x4;
typedef __attribute__((ext_vector_type(8))) unsigned int u32x8;

#define B_  4
#define T_  2048
#define D_  1024
#define H_  16
#define DH_ 64

__device__ __forceinline__ v8f wmma4(v2f a, v2f b, v8f c) {
    // 8 args: (neg_a, A, neg_b, B, c_mod, C, reuse_a, reuse_b)
    return __builtin_amdgcn_wmma_f32_16x16x4_f32(
        false, a, false, b, (short)0, c, false, false);
}

// ---------------------------------------------------------------------------
// TDM: DMA one 16-row x 64-float (16x256B) tile from global into LDS.
// D# per CDNA5 ISA:
//   group0: count=1 | lds_addr | global_addr[56:0] | type=2
//   group1: data_size=4B, tensor_dim0=64, tensor_dim1=T, tile=64x16,
//           tensor_dim0_stride=64 (all compile-time constant)
// ---------------------------------------------------------------------------
__device__ __forceinline__ void tdm_load_tile16x64(const float* gptr,
                                                   unsigned lds_off) {
    unsigned long long ga = (unsigned long long)(uintptr_t)gptr;
    u32x4 g0;
    g0[0] = 1u;                                   // count=1 (valid user D#)
    g0[1] = lds_off;                              // lds_addr (bytes)
    g0[2] = (unsigned)ga;                         // global_addr[31:0]
    g0[3] = ((unsigned)(ga >> 32) & 0x01FFFFFFu)  // global_addr[56:32]
            | (2u << 30);                         // type=2 ("image")
    u32x8 g1;
    g1[0] = 0x00020000u;                  // wg_mask=0, data_size=2 (4 bytes)
    g1[1] = (unsigned)(DH_ & 0xFFFF) << 16;           // tensor_dim0 = 64
    g1[2] = (unsigned)(DH_ >> 16) | ((T_ & 0xFFFF) << 16); // tensor_dim1 = 2048
    g1[3] = (unsigned)(T_ >> 16) | ((unsigned)DH_ << 16);  // tile_dim0 = 64
    g1[4] = 16u;                          // tile_dim1 = 16 rows, tile_dim2 = 0
    g1[5] = (unsigned)DH_;                // tensor_dim0_stride = 64 (lo 32)
    g1[6] = 0u;                           // stride0 hi, tensor_dim1_stride lo
    g1[7] = 0u;                           // tensor_dim1_stride hi
    asm volatile("tensor_load_to_lds %0, %1" :: "s"(g0), "s"(g1) : "memory");
}

// ---------------------------------------------------------------------------
// Kernel 1: qkv = x @ Wqkv^T   (M=8192, N=3072, K=1024)
// One 16x16 output tile per wave; scatter into [B,H,T,Dh] q/k/v buffers.
// ---------------------------------------------------------------------------
__global__ __launch_bounds__(256) void qkv_gemm_kernel(
    const float* __restrict__ x, const float* __restrict__ Wqkv,
    float* __restrict__ qws, float* __restrict__ kws, float* __restrict__ vws) {
    const int lane = threadIdx.x & 31;
    const int wave = threadIdx.x >> 5;
    const int tile = blockIdx.x * 8 + wave;
    const int NT = (3 * D_) / 16;              // 192 N-tiles
    const int tm = tile / NT, tn = tile % NT;
    const int m0 = tm * 16, n0 = tn * 16;
    const int r   = lane & 15;
    const int kk  = ((lane >> 4) & 1) * 2;     // A/B ISA layout: hi lanes K+2
    const int rhi = ((lane >> 4) & 1) * 8;

    const float* arow = x    + (size_t)(m0 + r) * D_ + kk;
    const float* brow = Wqkv + (size_t)(n0 + r) * D_ + kk;

    v8f c = {};
#pragma unroll 8
    for (int k0 = 0; k0 < D_; k0 += 4) {
        v2f a = *(const v2f*)(arow + k0);
        v2f b = *(const v2f*)(brow + k0);
        c = wmma4(a, b, c);
    }

    // epilogue: e = col -> (which of q/k/v, head, dh); row -> (b, t)
    const int col  = n0 + r;
    const int cidx = col >> 10;
    const int rem  = col & 1023;
    const int h    = rem >> 6;
    const int dh   = rem & 63;
    float* dst = (cidx == 0) ? qws : ((cidx == 1) ? kws : vws);
#pragma unroll
    for (int rr = 0; rr < 8; ++rr) {
        int mrow = m0 + rr + rhi;
        int bb = mrow >> 11;           // / T_
        int tt = mrow & (T_ - 1);
        dst[(((size_t)bb * H_ + h) * T_ + tt) * DH_ + dh] = c[rr];
    }
}

// ---------------------------------------------------------------------------
// Kernel 2: in-place RoPE on q and k. Each thread owns rotation pair (j, j+32).
// ---------------------------------------------------------------------------
__global__ __launch_bounds__(256) void rope_kernel(float* __restrict__ qws,
                                                   float* __restrict__ kws) {
    const long long idx = (long long)blockIdx.x * 256 + threadIdx.x;
    const int w = (int)(idx & 63);
    const long long row = idx >> 6;            // [0, B*H*T)
    const int t = (int)(row & (T_ - 1));
    const int j = w & 31;
    float* p = ((w < 32) ? qws : kws) + row * DH_;
    float x0 = p[j], x1 = p[j + 32];
    float inv = __powf(10000.0f, -(float)j * (1.0f / 32.0f));
    float ang = (float)t * inv;
    float s, co;
    __sincosf(ang, &s, &co);
    p[j]      = x0 * co - x1 * s;
    p[j + 32] = x1 * co + x0 * s;
}

// ---------------------------------------------------------------------------
// Kernel 3: flash-style attention per (b,h). 4 waves/block, each wave owns
// 16 q-rows x Dh=64. K/V tiles staged to LDS by the TDM (double-buffered,
// issued by wave 0, split-waited with s_wait_tensorcnt). Q held in registers.
// Online softmax via wave32 shfl reductions; P relaid through LDS for the
// PV WMMA A-operand.
// ---------------------------------------------------------------------------
__global__ __launch_bounds__(128) void attn_kernel(
    const float* __restrict__ qws, const float* __restrict__ kws,
    const float* __restrict__ vws, const unsigned char* __restrict__ mask,
    float* __restrict__ ctx) {
    __shared__ float Kt[2][16][DH_];
    __shared__ float Vt[2][16][DH_];
    __shared__ float Plds[4][16][17];

    const int lane = threadIdx.x & 31;
    const int wave = threadIdx.x >> 5;
    const int qb = blockIdx.x & 31;            // T/64 = 32 q-tiles
    const int bh = blockIdx.x >> 5;            // b*H + h
    const int b  = bh >> 4;
    const int h  = bh & 15;

    const float* Q = qws + (size_t)bh * T_ * DH_;
    const float* K = kws + (size_t)bh * T_ * DH_;
    const float* V = vws + (size_t)bh * T_ * DH_;
    const unsigned char* mrow = mask + (size_t)b * T_;

    const int q0  = qb * 64 + wave * 16;
    const int r   = lane & 15;
    const int kk  = ((lane >> 4) & 1) * 2;
    const int rhi = ((lane >> 4) & 1) * 8;

    // Hoist this wave's Q A-operand slice into registers (32 VGPRs).
    v2f qreg[16];
    {
        const float* qrow = Q + (size_t)(q0 + r) * DH_ + kk;
#pragma unroll
        for (int i = 0; i < 16; ++i) qreg[i] = *(const v2f*)(qrow + 4 * i);
    }

    v8f acc0 = {}, acc1 = {}, acc2 = {}, acc3 = {};
    float mstat[8], lstat[8];
#pragma unroll
    for (int i = 0; i < 8; ++i) { mstat[i] = -3.0e38f; lstat[i] = 0.0f; }

    const unsigned kt_lds0 = (unsigned)(uintptr_t)&Kt[0][0][0];
    const unsigned kt_lds1 = (unsigned)(uintptr_t)&Kt[1][0][0];
    const unsigned vt_lds0 = (unsigned)(uintptr_t)&Vt[0][0][0];
    const unsigned vt_lds1 = (unsigned)(uintptr_t)&Vt[1][0][0];

    const int NKT = T_ / 16;                   // 128 key tiles

    // Prologue: DMA tile 0 into buffer 0 (wave 0 only; TDM ignores EXEC but
    // we want a single wave's TENSORcnt tracking the transfers).
    if (wave == 0) {
        tdm_load_tile16x64(K, kt_lds0);
        tdm_load_tile16x64(V, vt_lds0);
    }

    for (int kt = 0; kt < NKT; ++kt) {
        const int kbase = kt * 16;
        const int buf = kt & 1;
        if (wave == 0) {
            if (kt + 1 < NKT) {
                const float* kn = K + (size_t)(kbase + 16) * DH_;
                const float* vn = V + (size_t)(kbase + 16) * DH_;
                tdm_load_tile16x64(kn, buf ? kt_lds0 : kt_lds1);
                tdm_load_tile16x64(vn, buf ? vt_lds0 : vt_lds1);
                // In-order TDM completion: <=2 outstanding => current buffer
                // has landed while the next buffer streams in.
                __builtin_amdgcn_s_wait_tensorcnt(2);
            } else {
                __builtin_amdgcn_s_wait_tensorcnt(0);
            }
        }
        __syncthreads();                       // release K/V tile to all waves

        // S = Q @ K^T : 16 WMMAs over Dh, K-operand from LDS
        v8f S = {};
        const float* krow = &Kt[buf][r][kk];
#pragma unroll
        for (int i = 0; i < 16; ++i) {
            v2f bb = *(const v2f*)(krow + 4 * i);
            S = wmma4(qreg[i], bb, S);
        }
        // scale + padding mask (lane <-> key in C layout), online softmax
        const float madd = mrow[kbase + r] ? -3.0e38f : 0.0f;
#pragma unroll
        for (int rr = 0; rr < 8; ++rr) {
            float s = S[rr] * 0.125f + madd;    // 1/sqrt(64)
            float rmax = s;
            rmax = fmaxf(rmax, __shfl_xor(rmax, 1, 32));
            rmax = fmaxf(rmax, __shfl_xor(rmax, 2, 32));
            rmax = fmaxf(rmax, __shfl_xor(rmax, 4, 32));
            rmax = fmaxf(rmax, __shfl_xor(rmax, 8, 32));
            float mnew = fmaxf(mstat[rr], rmax);
            float corr = __expf(mstat[rr] - mnew);
            float p = __expf(s - mnew);
            float rs = p;
            rs += __shfl_xor(rs, 1, 32);
            rs += __shfl_xor(rs, 2, 32);
            rs += __shfl_xor(rs, 4, 32);
            rs += __shfl_xor(rs, 8, 32);
            lstat[rr] = lstat[rr] * corr + rs;
            mstat[rr] = mnew;
            acc0[rr] *= corr; acc1[rr] *= corr;
            acc2[rr] *= corr; acc3[rr] *= corr;
            S[rr] = p;
        }
        // stage P (C layout -> LDS row-major) for PV A-operand
#pragma unroll
        for (int rr = 0; rr < 8; ++rr) Plds[wave][rr + rhi][r] = S[rr];
        // O += P @ V : K-dim is the 16 keys, split into 4 WMMAs x 4 d-tiles;
        // V-operand from LDS. (Plds is private per wave; LDS ops from one
        // wave complete in order, no cross-wave sync needed here.)
#pragma unroll
        for (int kc = 0; kc < 16; kc += 4) {
            v2f a;
            a.x = Plds[wave][r][kc + kk];
            a.y = Plds[wave][r][kc + kk + 1];
            const float* v0 = &Vt[buf][kc + kk][0];
            const float* v1 = &Vt[buf][kc + kk + 1][0];
            v2f b0; b0.x = v0[r];      b0.y = v1[r];
            v2f b1; b1.x = v0[16 + r]; b1.y = v1[16 + r];
            v2f b2; b2.x = v0[32 + r]; b2.y = v1[32 + r];
            v2f b3; b3.x = v0[48 + r]; b3.y = v1[48 + r];
            acc0 = wmma4(a, b0, acc0);
            acc1 = wmma4(a, b1, acc1);
            acc2 = wmma4(a, b2, acc2);
            acc3 = wmma4(a, b3, acc3);
        }
        __syncthreads();   // all waves done reading buf before TDM overwrites
    }
    // epilogue: ctx layout [B][T][H][Dh] so the out-proj reads contiguous K
#pragma unroll
    for (int rr = 0; rr < 8; ++rr) {
        int t = q0 + rr + rhi;
        float invl = 1.0f / lstat[rr];
        size_t base = (((size_t)(b * T_ + t)) * H_ + h) * DH_;
        ctx[base +  0 + r] = acc0[rr] * invl;
        ctx[base + 16 + r] = acc1[rr] * invl;
        ctx[base + 32 + r] = acc2[rr] * invl;
        ctx[base + 48 + r] = acc3[rr] * invl;
    }
}

// ---------------------------------------------------------------------------
// Kernel 4: out = ctx @ Wo^T + bo   (M=8192, N=1024, K=1024)
// ---------------------------------------------------------------------------
__global__ __launch_bounds__(256) void out_gemm_kernel(
    const float* __restrict__ ctx, const float* __restrict__ Wo,
    const float* __restrict__ bo, float* __restrict__ out) {
    const int lane = threadIdx.x & 31;
    const int wave = threadIdx.x >> 5;
    const int tile = blockIdx.x * 8 + wave;
    const int NT = D_ / 16;                    // 64
    const int tm = tile / NT, tn = tile % NT;
    const int m0 = tm * 16, n0 = tn * 16;
    const int r   = lane & 15;
    const int kk  = ((lane >> 4) & 1) * 2;
    const int rhi = ((lane >> 4) & 1) * 8;

    const float* arow = ctx + (size_t)(m0 + r) * D_ + kk;
    const float* brow = Wo  + (size_t)(n0 + r) * D_ + kk;

    v8f c = {};
#pragma unroll 8
    for (int k0 = 0; k0 < D_; k0 += 4) {
        v2f a = *(const v2f*)(arow + k0);
        v2f b = *(const v2f*)(brow + k0);
        c = wmma4(a, b, c);
    }
    const int col = n0 + r;
    const float bias = bo[col];
#pragma unroll
    for (int rr = 0; rr < 8; ++rr) {
        int mrow = m0 + rr + rhi;
        out[(size_t)mrow * D_ + col] = c[rr] + bias;
    }
}

// ---------------------------------------------------------------------------
extern "C" void kernel_launch(void* const* d_in, const int* in_sizes, int n_in,
                              void* d_out, int out_size, void* d_ws, size_t ws_size,
                              hipStream_t stream) {
    const float*         x    = (const float*)d_in[0];
    const unsigned char* mask = (const unsigned char*)d_in[1];
    const float*         Wqkv = (const float*)d_in[2];
    const float*         Wo   = (const float*)d_in[3];
    const float*         bo   = (const float*)d_in[4];
    float* out = (float*)d_out;
    float* ws  = (float*)d_ws;

    const size_t per = (size_t)B_ * H_ * T_ * DH_;   // 8,388,608 floats (32 MB)
    float* qws = ws;
    float* kws = ws + per;
    float* vws = ws + 2 * per;
    float* ctx = ws + 3 * per;

    // QKV: 512*192 = 98304 wave-tiles / 8 waves per block
    qkv_gemm_kernel<<<12288, 256, 0, stream>>>(x, Wqkv, qws, kws, vws);
    // RoPE: B*H*T rows * 64 threads (32 q-pairs + 32 k-pairs)
    rope_kernel<<<32768, 256, 0, stream>>>(qws, kws);
    // Attention: (B*H) * (T/64) blocks of 4 waves
    attn_kernel<<<2048, 128, 0, stream>>>(qws, kws, vws, mask, ctx);
    // Output projection: 512*64 = 32768 wave-tiles / 8 per block
    out_gemm_kernel<<<4096, 256, 0, stream>>>(ctx, Wo, bo, out);
}